// SimpleRecurrentLayer_10333691314205
// MI455X (gfx1250) — compile-verified
//
#include <hip/hip_runtime.h>
#include <cmath>

// GRU: T=512 steps, B=128, D=H=512, gates G=3H=1536.
// Phase 1: x_proj GEMM [65536 x 512] x [512 x 1536]  (parallel, WMMA f32)
// Phase 2: persistent recurrent kernel, 32 WGs x 8 waves.
//   WG = one hidden block (16 cols), wave = one batch block (16 rows).
//   Each WG stages its 3-gate W_hh slice (96 KB) into LDS once, pre-swizzled
//   into the V_WMMA_F32_16X16X4_F32 B-operand lane layout; all 512 timesteps
//   then read weights with ds_load_b64 instead of hitting L2.
// Uses V_WMMA_F32_16X16X4_F32 to match the fp32 reference numerics.

typedef float v2f __attribute__((ext_vector_type(2)));
typedef float v8f __attribute__((ext_vector_type(8)));

#define T_DIM 512
#define B_DIM 128
#define D_DIM 512
#define H_DIM 512
#define G_DIM (3 * H_DIM)   // 1536
#define THREADS 256
#define NWG_REC 32          // 32 WGs (one per hidden block) * 8 waves (batch blocks)

// LDS weight stage: 3 gates * 128 K-steps * 32 lane-slots of float2 = 96 KB
#define KSTEPS (H_DIM / 4)                   // 128
#define LDS_W_ENTRIES (3 * KSTEPS * 32)      // 12288 v2f entries
#define LDS_W_BYTES (LDS_W_ENTRIES * 8)      // 98304 bytes

__device__ __forceinline__ v8f wmma_f32(v2f a, v2f b, v8f c) {
  // (neg_a, A, neg_b, B, c_mod, C, reuse_a, reuse_b)
  return __builtin_amdgcn_wmma_f32_16x16x4_f32(false, a, false, b,
                                               (short)0, c, false, false);
}

// ---------------------------------------------------------------------------
// Init: zero h0 and the grid-barrier words (workspace is poisoned; reset every
// call so graph replays are deterministic).
// ---------------------------------------------------------------------------
__global__ void gru_init(float* __restrict__ hbuf, unsigned* __restrict__ bar) {
  for (int i = threadIdx.x; i < B_DIM * H_DIM; i += THREADS) hbuf[i] = 0.0f;
  if (threadIdx.x == 0) { bar[0] = 0u; bar[1] = 0u; }
}

// ---------------------------------------------------------------------------
// Phase 1: x_proj[m, g] = sum_d inputs[m, d] * W_ih[g, d] + b_ih[g]
// Each wave: 4 M-tiles (64 rows) x 1 N-tile (16 cols), K = 512 in steps of 4.
// Waves within a WG share the N-block -> weight float2 loads hit L0.
// ---------------------------------------------------------------------------
__global__ void __launch_bounds__(THREADS) xproj_gemm(
    const float* __restrict__ x, const float* __restrict__ Wih,
    const float* __restrict__ bih, float* __restrict__ xp) {
  const int lane = threadIdx.x & 31;
  const int wid  = blockIdx.x * (THREADS / 32) + (threadIdx.x >> 5);
  const int nb = wid >> 10;          // 0..95   (1024 waves share one N-block)
  const int mg = wid & 1023;         // 0..1023 (64 rows each)
  const int n0 = nb * 16;
  const int m0 = mg * 64;
  const int lm   = lane & 15;
  const int koff = (lane >> 4) << 1; // K pair base per ISA A-layout

  const float* a0 = x   + (size_t)(m0 + lm) * D_DIM + koff;
  const float* bp = Wih + (size_t)(n0 + lm) * D_DIM + koff;

  v8f acc0 = {}, acc1 = {}, acc2 = {}, acc3 = {};
  for (int k = 0; k < D_DIM; k += 4) {
    v2f b   = *(const v2f*)(bp + k);
    v2f aa0 = *(const v2f*)(a0 + k);
    v2f aa1 = *(const v2f*)(a0 + 16 * D_DIM + k);
    v2f aa2 = *(const v2f*)(a0 + 32 * D_DIM + k);
    v2f aa3 = *(const v2f*)(a0 + 48 * D_DIM + k);
    acc0 = wmma_f32(aa0, b, acc0);
    acc1 = wmma_f32(aa1, b, acc1);
    acc2 = wmma_f32(aa2, b, acc2);
    acc3 = wmma_f32(aa3, b, acc3);
  }

  const float bias = bih[n0 + lm];
  const int   col  = n0 + lm;
  const int rbase  = m0 + ((lane >> 4) << 3); // C/D: M = i + 8*(lane/16)
#pragma unroll
  for (int i = 0; i < 8; ++i) {
    xp[(size_t)(rbase +      i) * G_DIM + col] = acc0[i] + bias;
    xp[(size_t)(rbase + 16 + i) * G_DIM + col] = acc1[i] + bias;
    xp[(size_t)(rbase + 32 + i) * G_DIM + col] = acc2[i] + bias;
    xp[(size_t)(rbase + 48 + i) * G_DIM + col] = acc3[i] + bias;
  }
}

// ---------------------------------------------------------------------------
// Phase 2: persistent recurrent scan.
//   blockIdx.x = hj (hidden block, 0..31); wave id = bi (batch block, 0..7).
//   W_hh slice for this hj (3 gates x 16 rows x 512 K) staged in LDS once,
//   in B-operand lane layout: entry[gate][kstep][lane] = float2.
//   h (256 KB) stays hot in the 192 MB L2; xp streamed once per step.
// ---------------------------------------------------------------------------
__global__ void __launch_bounds__(THREADS, 1) gru_scan(
    const float* __restrict__ Whh, const float* __restrict__ bhh,
    const float* __restrict__ xp, float* __restrict__ hbuf,
    float* __restrict__ out, unsigned* __restrict__ bar) {
  extern __shared__ v2f lds_w[];   // LDS_W_ENTRIES (96 KB), dynamic

  const int lane = threadIdx.x & 31;
  const int bi   = threadIdx.x >> 5;   // batch block 0..7
  const int hj   = blockIdx.x;         // hidden block 0..31
  const int b0 = bi * 16, c0 = hj * 16;
  const int lm   = lane & 15;
  const int koff = (lane >> 4) << 1;
  const int col  = c0 + lm;

  // ---- cooperative stage of W_hh into LDS, pre-swizzled for WMMA B ----
  // gh[b, g] = sum_k h[b, k] * W_hh[g, k] -> B[k, n] = W_hh[gate*H + c0 + n, k]
  for (int e = threadIdx.x; e < LDS_W_ENTRIES; e += THREADS) {
    const int gate  = e >> 12;          // e / (KSTEPS*32)
    const int rem   = e & 4095;
    const int kstep = rem >> 5;
    const int ls    = rem & 31;
    const int row   = gate * H_DIM + c0 + (ls & 15);
    const int kk    = kstep * 4 + ((ls >> 4) << 1);
    lds_w[e] = *(const v2f*)(Whh + (size_t)row * H_DIM + kk);
  }
  __syncthreads();

  const v2f* wr = lds_w + 0 * (KSTEPS * 32) + lane;
  const v2f* wz = lds_w + 1 * (KSTEPS * 32) + lane;
  const v2f* wn = lds_w + 2 * (KSTEPS * 32) + lane;
  const float br = bhh[col], bz = bhh[H_DIM + col], bn = bhh[2 * H_DIM + col];

  for (int t = 0; t < T_DIM; ++t) {
    const float* hprev = hbuf + (size_t)(t & 1) * (B_DIM * H_DIM);
    float*       hnext = hbuf + (size_t)((t + 1) & 1) * (B_DIM * H_DIM);
    const float* ap = hprev + (size_t)(b0 + lm) * H_DIM + koff;

    v8f ar = {}, az = {}, an = {};
    for (int ks = 0; ks < KSTEPS; ++ks) {
      v2f a   = *(const v2f*)(ap + ks * 4);
      v2f wbr = wr[ks * 32];
      v2f wbz = wz[ks * 32];
      v2f wbn = wn[ks * 32];
      ar = wmma_f32(a, wbr, ar);
      az = wmma_f32(a, wbz, az);
      an = wmma_f32(a, wbn, an);
    }

    const float* xpt   = xp + (size_t)t * (B_DIM * G_DIM);
    const int    rbase = b0 + ((lane >> 4) << 3);
#pragma unroll
    for (int i = 0; i < 8; ++i) {
      const int row = rbase + i;
      const float* xrow = xpt + (size_t)row * G_DIM;
      const float xr = xrow[col];
      const float xz = xrow[H_DIM + col];
      const float xn = xrow[2 * H_DIM + col];
      const float hp = hprev[(size_t)row * H_DIM + col];
      const float r  = 1.0f / (1.0f + __expf(-(xr + ar[i] + br)));
      const float z  = 1.0f / (1.0f + __expf(-(xz + az[i] + bz)));
      const float nn = tanhf(xn + r * (an[i] + bn));
      const float hnew = (1.0f - z) * nn + z * hp;
      hnext[(size_t)row * H_DIM + col] = hnew;
      out[(size_t)t * (B_DIM * H_DIM) + (size_t)row * H_DIM + col] = hnew;
    }

    // ---- device-scope barrier between timesteps ----
    __threadfence();
    __syncthreads();
    if (threadIdx.x == 0) {
      if (atomicAdd(&bar[0], 1u) == (unsigned)(NWG_REC - 1)) {
        atomicExch(&bar[0], 0u);   // reset arrival count
        __threadfence();
        atomicAdd(&bar[1], 1u);    // release generation t+1
      } else {
        while (__hip_atomic_load(&bar[1], __ATOMIC_ACQUIRE,
                                 __HIP_MEMORY_SCOPE_AGENT) < (unsigned)(t + 1)) {
          __builtin_amdgcn_s_sleep(2);
        }
      }
    }
    __syncthreads();
  }
}

// ---------------------------------------------------------------------------
// inputs order: inputs, W_ih, W_hh, b_ih, b_hh
// workspace: [bar: 256 B][hbuf: 2*B*H f32 = 512 KB][xp: T*B*3H f32 = 384 MB]
// ---------------------------------------------------------------------------
extern "C" void kernel_launch(void* const* d_in, const int* in_sizes, int n_in,
                              void* d_out, int out_size, void* d_ws, size_t ws_size,
                              hipStream_t stream) {
  const float* x   = (const float*)d_in[0];
  const float* Wih = (const float*)d_in[1];
  const float* Whh = (const float*)d_in[2];
  const float* bih = (const float*)d_in[3];
  const float* bhh = (const float*)d_in[4];
  float* out = (float*)d_out;

  unsigned* bar  = (unsigned*)d_ws;
  float*    hbuf = (float*)((char*)d_ws + 256);
  float*    xp   = hbuf + 2 * B_DIM * H_DIM;

  gru_init<<<1, THREADS, 0, stream>>>(hbuf, bar);

  // 96 N-blocks * 1024 M-groups = 98304 waves / 8 waves-per-WG = 12288 WGs
  xproj_gemm<<<12288, THREADS, 0, stream>>>(x, Wih, bih, xp);

  gru_scan<<<NWG_REC, THREADS, LDS_W_BYTES, stream>>>(Whh, bhh, xp, hbuf, out, bar);
}